// ModularNet_81054622810212
// MI455X (gfx1250) — compile-verified
//
#include <hip/hip_runtime.h>
#include <hip/hip_bf16.h>

// ---------------------------------------------------------------------------
// ModularNet on gfx1250 (MI455X):
//   out0: y (16,128,32,32) f32   -- two selected-expert 3x3 convs + ReLU
//   out1: ctl (16,128,2) f32
//   out2: ctl_nearest (16,128,2) f32
// Conv is matrix-core bound (~9.7 GFLOP, all data L2-resident) -> implicit
// GEMM with V_WMMA_F32_16X16X4_F32 (full f32 path). Weights pre-transposed
// to [expert][co][tap][ci] (A-frag = one global_load_b64/lane); LDS patch
// stored [pos][ci] with padded stride so B-frag = one ds_load_b64/lane.
// Patch fill uses CDNA5 async global->LDS loads (ASYNCcnt).
// ---------------------------------------------------------------------------

#define N_MOD  8
#define DEPTH  2
#define DIM_E  128
#define C_CH   128
#define H_SP   32
#define W_SP   32
#define B_N    16
#define HWSZ   (H_SP * W_SP)          // 1024
#define CHW    (C_CH * HWSZ)          // 131072
#define WEXP_STRIDE (C_CH * C_CH * 9) // 147456 floats per expert
#define KTOT   (C_CH * 9)             // 1152

typedef float v2f __attribute__((ext_vector_type(2)));
typedef float v8f __attribute__((ext_vector_type(8)));

// ---------------------------------------------------------------------------
// Kernel 1: pooled -> ctl -> argmin expert ids -> ctl / ctl_nearest outputs.
// ---------------------------------------------------------------------------
__global__ __launch_bounds__(128)
void ctl_kernel(const float* __restrict__ x,
                const float* __restrict__ W_ctl,
                const float* __restrict__ b_ctl,
                const float* __restrict__ emb,
                float* __restrict__ out_ctl,      // (B, DIM_E, DEPTH) flat
                float* __restrict__ out_ctln,     // (B, DIM_E, DEPTH) flat
                int* __restrict__ idx_ws)         // (B, DEPTH)
{
    __shared__ float s_pooled[C_CH];
    __shared__ float s_ctl[DEPTH * DIM_E];
    __shared__ float s_d2[DEPTH * N_MOD];
    __shared__ int   s_idx[DEPTH];

    const int b = blockIdx.x;
    const int c = threadIdx.x;            // 0..127

    const float* px = x + (size_t)b * CHW + (size_t)c * HWSZ;
    float acc = 0.f;
    #pragma unroll 4
    for (int i = 0; i < HWSZ; ++i) acc += px[i];
    s_pooled[c] = acc * (1.0f / (float)HWSZ);
    __syncthreads();

    for (int j = c; j < DEPTH * DIM_E; j += 128) {
        const float* wr = W_ctl + (size_t)j * C_CH;
        float a = b_ctl[j];
        #pragma unroll 8
        for (int k = 0; k < C_CH; ++k) a += s_pooled[k] * wr[k];
        s_ctl[j] = a;
        out_ctl[(size_t)b * (DEPTH * DIM_E) + j] = a;
    }
    __syncthreads();

    if (c < DEPTH * N_MOD) {
        const int t = c >> 3;
        const int m = c & 7;
        float d = 0.f;
        #pragma unroll 8
        for (int e = 0; e < DIM_E; ++e) {
            float diff = s_ctl[e * DEPTH + t] - emb[e * N_MOD + m];
            d += diff * diff;
        }
        s_d2[c] = d;
    }
    __syncthreads();

    if (c < DEPTH) {
        const int t = c;
        float best = s_d2[t * N_MOD];
        int bi = 0;
        #pragma unroll
        for (int m = 1; m < N_MOD; ++m) {
            float v = s_d2[t * N_MOD + m];
            if (v < best) { best = v; bi = m; }   // first-min = jnp.argmin
        }
        s_idx[t] = bi;
        idx_ws[b * DEPTH + t] = bi;
    }
    __syncthreads();

    {
        const int e = c;
        #pragma unroll
        for (int t = 0; t < DEPTH; ++t)
            out_ctln[(size_t)b * (DEPTH * DIM_E) + e * DEPTH + t] =
                emb[e * N_MOD + s_idx[t]];
    }
}

// ---------------------------------------------------------------------------
// Kernel 2: weight transpose prep.
//   W_exp[mod][co][ci][tap]  ->  Wt[mod][co][tap][ci]   (ci contiguous)
// ---------------------------------------------------------------------------
#define WT_TOTAL (N_MOD * C_CH * 9 * C_CH)   // 1,179,648

__global__ __launch_bounds__(256)
void wexp_transpose_kernel(const float* __restrict__ W_exp,
                           float* __restrict__ Wt)
{
    const int idx = blockIdx.x * 256 + threadIdx.x;
    if (idx >= WT_TOTAL) return;
    const int ci  = idx & (C_CH - 1);
    const int mt  = idx >> 7;        // mc*9 + tap
    const int tap = mt % 9;
    const int mc  = mt / 9;          // mod*128 + co
    Wt[idx] = W_exp[(size_t)mc * KTOT + ci * 9 + tap];
}

// ---------------------------------------------------------------------------
// Kernel 3: selected-expert 3x3 conv + bias + ReLU via implicit GEMM WMMA.
//   grid = (32 image rows, 16 batches), block = 256 threads (8 waves).
//   LDS patch layout: [pos][ci], pos = row*34+col (3x34 haloed positions),
//   ci stride padded to 132 (conflict-free ds_load_b64 across 32 lanes).
//   Wave w: 32(co) x 32(pix) tile; per K-step of 4:
//     2x global_load_b64 (A) + 2x ds_load_b64 (B) -> 4x wmma_f32_16x16x4.
// A 16x4 f32 layout (ISA 7.12.2): lanes 0-15 -> {K0,K1}, lanes 16-31 ->
// {K2,K3}; B 4x16 mirrored.
// ---------------------------------------------------------------------------
#define PATCH_COLS 34
#define PPOS       (3 * PATCH_COLS)          // 102 spatial positions
#define PSTRIDE    132                       // padded ci stride (floats)
#define PATCH_ELEMS (PPOS * PSTRIDE)         // 13464 floats = 53,856 B

__global__ __launch_bounds__(256)
void conv_expert_kernel(const float* __restrict__ in,    // (B,C,H,W)
                        const float* __restrict__ Wt,    // (N_MOD,C,9,C)
                        const float* __restrict__ b_exp, // (N_MOD,C)
                        const int* __restrict__ idx_ws,  // (B,DEPTH)
                        const int t,
                        float* __restrict__ out)         // (B,C,H,W)
{
    __shared__ float s_patch[PATCH_ELEMS];

    const int h0 = blockIdx.x;           // image row 0..31
    const int b  = blockIdx.y;           // batch

    const int expert = idx_ws[b * DEPTH + t];
    const float* wbase_e = Wt + (size_t)expert * WEXP_STRIDE;
    const float* bias    = b_exp + (size_t)expert * C_CH;

    // ---- haloed input row patch -> LDS (transposed to [pos][ci]) ----
    // In-bounds elements: CDNA5 async global->LDS copy (ASYNCcnt).
    // Halo / out-of-bounds: plain zero stores (disjoint addresses).
    const float* inb = in + (size_t)b * CHW;
    for (int e = threadIdx.x; e < C_CH * PPOS; e += 256) {
        const int ci  = e / PPOS;
        const int rem = e - ci * PPOS;          // pos = row*34 + col
        const int row = rem / PATCH_COLS;
        const int col = rem - row * PATCH_COLS;
        const int hin = h0 + row - 1;
        const int win = col - 1;
        float* dstp = &s_patch[rem * PSTRIDE + ci];
        if (hin >= 0 && hin < H_SP && win >= 0 && win < W_SP) {
            const float* gsrc = &inb[(size_t)ci * HWSZ + hin * W_SP + win];
            const unsigned ldsoff = (unsigned)(unsigned long long)dstp;
            asm volatile("global_load_async_to_lds_b32 %0, %1, off"
                         :: "v"(ldsoff), "v"(gsrc) : "memory");
        } else {
            *dstp = 0.0f;
        }
    }
    asm volatile("s_wait_asynccnt 0x0" ::: "memory");
    __syncthreads();

    // ---- per-wave implicit GEMM: M=32 (co), N=32 (pixels), K=1152 ----
    const int wave    = threadIdx.x >> 5;     // 0..7
    const int lane    = threadIdx.x & 31;
    const int halfsel = lane >> 4;            // 0: K0,K1 ; 1: K2,K3
    const int mn      = lane & 15;            // M for A-frag, N for B/D-frag
    const int co0     = wave * 32;

    const float* wr0 = wbase_e + (size_t)(co0 + mn)      * KTOT;
    const float* wr1 = wbase_e + (size_t)(co0 + 16 + mn) * KTOT;

    v8f acc00 = {}, acc01 = {}, acc10 = {}, acc11 = {};

    #pragma unroll
    for (int tap = 0; tap < 9; ++tap) {
        const int dh    = tap / 3 - 1;
        const int dw    = tap - (tap / 3) * 3 - 1;
        // spatial position of pixel n (tile0) for this tap
        const int pofs0 = ((1 + dh) * PATCH_COLS + (1 + mn + dw)) * PSTRIDE;
        const int pofs1 = pofs0 + 16 * PSTRIDE;          // n-tile 1 (+16 cols)
        const float* wa0 = wr0 + tap * C_CH;
        const float* wa1 = wr1 + tap * C_CH;

        for (int ci0 = 0; ci0 < C_CH; ci0 += 4) {
            const int ciA = ci0 + 2 * halfsel;
            const v2f a0 = *(const v2f*)(wa0 + ciA);          // global b64
            const v2f a1 = *(const v2f*)(wa1 + ciA);          // global b64
            const v2f b0 = *(const v2f*)&s_patch[pofs0 + ciA]; // ds b64
            const v2f b1 = *(const v2f*)&s_patch[pofs1 + ciA]; // ds b64
            acc00 = __builtin_amdgcn_wmma_f32_16x16x4_f32(
                false, a0, false, b0, (short)0, acc00, false, false);
            acc01 = __builtin_amdgcn_wmma_f32_16x16x4_f32(
                false, a0, false, b1, (short)0, acc01, false, false);
            acc10 = __builtin_amdgcn_wmma_f32_16x16x4_f32(
                false, a1, false, b0, (short)0, acc10, false, false);
            acc11 = __builtin_amdgcn_wmma_f32_16x16x4_f32(
                false, a1, false, b1, (short)0, acc11, false, false);
        }
    }

    // ---- epilogue: bias + ReLU + store ----
    float* outr = out + (size_t)b * CHW + h0 * W_SP;
    #pragma unroll
    for (int r = 0; r < 8; ++r) {
        const int coA = co0 + r + 8 * halfsel;
        const int coB = coA + 16;
        const float bA = bias[coA];
        const float bB = bias[coB];
        float v00 = fmaxf(acc00[r] + bA, 0.0f);
        float v01 = fmaxf(acc01[r] + bA, 0.0f);
        float v10 = fmaxf(acc10[r] + bB, 0.0f);
        float v11 = fmaxf(acc11[r] + bB, 0.0f);
        outr[(size_t)coA * HWSZ + mn]      = v00;
        outr[(size_t)coA * HWSZ + 16 + mn] = v01;
        outr[(size_t)coB * HWSZ + mn]      = v10;
        outr[(size_t)coB * HWSZ + 16 + mn] = v11;
    }
}

// ---------------------------------------------------------------------------
// Launch
// ---------------------------------------------------------------------------
extern "C" void kernel_launch(void* const* d_in, const int* in_sizes, int n_in,
                              void* d_out, int out_size, void* d_ws, size_t ws_size,
                              hipStream_t stream) {
    const float* x     = (const float*)d_in[0];
    const float* W_ctl = (const float*)d_in[1];
    const float* b_ctl = (const float*)d_in[2];
    const float* emb   = (const float*)d_in[3];
    const float* W_exp = (const float*)d_in[4];
    const float* b_exp = (const float*)d_in[5];

    float* out_y    = (float*)d_out;                         // 16*128*32*32
    float* out_ctl  = out_y + (size_t)B_N * CHW;             // 16*128*2
    float* out_ctln = out_ctl + (size_t)B_N * DEPTH * DIM_E; // 16*128*2

    int*   idx_ws = (int*)d_ws;                                   // 1KB
    float* Wt     = (float*)((char*)d_ws + 1024);                 // 4.5MB
    float* y_mid  = (float*)((char*)d_ws + 1024 +
                             (size_t)WT_TOTAL * sizeof(float));   // 8MB

    ctl_kernel<<<B_N, 128, 0, stream>>>(x, W_ctl, b_ctl, emb,
                                        out_ctl, out_ctln, idx_ws);

    wexp_transpose_kernel<<<(WT_TOTAL + 255) / 256, 256, 0, stream>>>(W_exp, Wt);

    dim3 grid(H_SP, B_N);
    conv_expert_kernel<<<grid, 256, 0, stream>>>(x,     Wt, b_exp, idx_ws, 0, y_mid);
    conv_expert_kernel<<<grid, 256, 0, stream>>>(y_mid, Wt, b_exp, idx_ws, 1, out_y);
}